// Model_23467701306048
// MI455X (gfx1250) — compile-verified
//
#include <hip/hip_runtime.h>
#include <math.h>

typedef float v2f __attribute__((ext_vector_type(2)));
typedef float v8f __attribute__((ext_vector_type(8)));

#define DIM   768
#define HDIM  64
#define NHEAD 12
#define BB    32
#define SS    64
#define NNS   65

__device__ __forceinline__ float wred_sum(float v) {
#pragma unroll
  for (int o = 16; o > 0; o >>= 1) v += __shfl_xor(v, o, 32);
  return v;
}
__device__ __forceinline__ float wred_max(float v) {
#pragma unroll
  for (int o = 16; o > 0; o >>= 1) v = fmaxf(v, __shfl_xor(v, o, 32));
  return v;
}

// ---------------------------------------------------------------------------
// Y[m, n] = sum_k X[m, k] * W[n*ldw + k]  (+ bias[n])
// f32 WMMA 16x16x4. One wave -> 16(M) x 64(N) tile, 8 waves/block -> 128 x 64.
// A frag (16x4 f32): lanes 0-15 hold M=lane, K=kk..kk+1; lanes 16-31 K=kk+2..3.
// B frag (4x16 f32): lanes 0-15 hold N=lane, K=kk..kk+1; lanes 16-31 K=kk+2..3.
// Out-of-range M rows are clamped to row 0 for loads (always valid memory);
// their accumulators are garbage but never stored (store guarded by row < M).
// ---------------------------------------------------------------------------
__global__ __launch_bounds__(256)
void gemm_wmma_f32(const float* __restrict__ X, const float* __restrict__ W,
                   const float* __restrict__ bias, float* __restrict__ Y,
                   int M, int ldw)
{
  const int lane = threadIdx.x & 31;
  const int wave = threadIdx.x >> 5;
  const int half = lane >> 4;     // 0: K0..1, 1: K2..3
  const int l16  = lane & 15;
  const int m0 = blockIdx.x * 128 + wave * 16;
  const int n0 = blockIdx.y * 64;

  const int arow = m0 + l16;
  const int arow_c = arow < M ? arow : 0;     // clamp, load unconditionally
  const float* aptr  = X + (size_t)arow_c * DIM + half * 2;
  const float* bptr  = W + (size_t)(n0 + l16) * (size_t)ldw + half * 2;
  const size_t bstep = (size_t)16 * (size_t)ldw;

  v8f acc0 = (v8f)0.0f, acc1 = (v8f)0.0f, acc2 = (v8f)0.0f, acc3 = (v8f)0.0f;

#pragma unroll 4
  for (int kk = 0; kk < DIM; kk += 4) {
    v2f a  = *(const v2f*)(aptr + kk);
    v2f b0 = *(const v2f*)(bptr + kk);
    v2f b1 = *(const v2f*)(bptr + bstep + kk);
    v2f b2 = *(const v2f*)(bptr + 2 * bstep + kk);
    v2f b3 = *(const v2f*)(bptr + 3 * bstep + kk);
    acc0 = __builtin_amdgcn_wmma_f32_16x16x4_f32(false, a, false, b0, (short)0, acc0, false, false);
    acc1 = __builtin_amdgcn_wmma_f32_16x16x4_f32(false, a, false, b1, (short)0, acc1, false, false);
    acc2 = __builtin_amdgcn_wmma_f32_16x16x4_f32(false, a, false, b2, (short)0, acc2, false, false);
    acc3 = __builtin_amdgcn_wmma_f32_16x16x4_f32(false, a, false, b3, (short)0, acc3, false, false);
  }

  // C/D layout: VGPR r -> M = m0 + r + (lane>=16 ? 8 : 0), N = n0 + (lane&15).
#pragma unroll
  for (int r = 0; r < 8; ++r) {
    const int row = m0 + r + half * 8;
    if (row < M) {
      const int col = n0 + l16;
      float b0 = bias ? bias[col     ] : 0.f;
      float b1 = bias ? bias[col + 16] : 0.f;
      float b2 = bias ? bias[col + 32] : 0.f;
      float b3 = bias ? bias[col + 48] : 0.f;
      float* yr = Y + (size_t)row * DIM;
      yr[col     ] = acc0[r] + b0;
      yr[col + 16] = acc1[r] + b1;
      yr[col + 32] = acc2[r] + b2;
      yr[col + 48] = acc3[r] + b3;
    }
  }
}

// ---------------------------------------------------------------------------
// W2e[h][d] = sum_{d'<64} we[d'] * eu_w2[(h*64+d')*768 + d];  bias2[h] = we.b2[h]
// ---------------------------------------------------------------------------
__global__ __launch_bounds__(256)
void prep_w2e(const float* __restrict__ eu_w2, const float* __restrict__ eu_b2,
              const float* __restrict__ attn_w, float* __restrict__ W2e,
              float* __restrict__ bias2)
{
  const int idx = blockIdx.x * 256 + threadIdx.x;
  if (idx < NHEAD * DIM) {
    const int h = idx / DIM, d = idx % DIM;
    float acc = 0.f;
#pragma unroll 4
    for (int dp = 0; dp < HDIM; ++dp)
      acc += attn_w[2 * HDIM + dp] * eu_w2[(size_t)(h * HDIM + dp) * DIM + d];
    W2e[idx] = acc;
  }
  if (idx < NHEAD) {
    float acc = 0.f;
    for (int dp = 0; dp < HDIM; ++dp)
      acc += attn_w[2 * HDIM + dp] * eu_b2[idx * HDIM + dp];
    bias2[idx] = acc;
  }
}

// ---------------------------------------------------------------------------
// qs[(b*H+h)*NS+n] = Q[b,n,h*64:].wq ; ks likewise with wk.
// ---------------------------------------------------------------------------
__global__ __launch_bounds__(256)
void qk_scores(const float* __restrict__ Q, const float* __restrict__ K,
               const float* __restrict__ attn_w,
               float* __restrict__ qs, float* __restrict__ ks)
{
  const int idx = blockIdx.x * 256 + threadIdx.x;
  if (idx >= BB * NHEAD * NNS) return;
  const int n = idx % NNS;
  const int h = (idx / NNS) % NHEAD;
  const int b = idx / (NNS * NHEAD);
  const float* qrow = Q + ((size_t)b * NNS + n) * DIM + h * HDIM;
  const float* krow = K + ((size_t)b * NNS + n) * DIM + h * HDIM;
  float aq = 0.f, ak = 0.f;
#pragma unroll 4
  for (int d = 0; d < HDIM; ++d) {
    aq += qrow[d] * attn_w[d];
    ak += krow[d] * attn_w[HDIM + d];
  }
  qs[idx] = aq;
  ks[idx] = ak;
}

// ---------------------------------------------------------------------------
// One block per (row i, batch b): edge LN+relu+head-dot, softmax, attn@V.
// x(i,j) = A[src] + Bv[j], src = (i==0 ? j : i).  Masked: j==0, j==i (i>=1).
// ---------------------------------------------------------------------------
__global__ __launch_bounds__(256)
void edge_attn_ctx(const float* __restrict__ Ab, const float* __restrict__ Bb,
                   const float* __restrict__ Vb,
                   const float* __restrict__ lng, const float* __restrict__ lnb,
                   const float* __restrict__ W2e, const float* __restrict__ bias2,
                   const float* __restrict__ qs, const float* __restrict__ ks,
                   const float* __restrict__ attn_b,
                   float* __restrict__ attn_out, float* __restrict__ ctx)
{
  __shared__ float sW[NHEAD * DIM];   // 36 KB
  __shared__ float sg[DIM];
  __shared__ float sb[DIM];
  __shared__ float sa[DIM];
  __shared__ float ssc[NHEAD][NNS];
  __shared__ float sp[NHEAD][NNS];

  const int i    = blockIdx.x;        // 0..64
  const int b    = blockIdx.y;        // 0..31
  const int tid  = threadIdx.x;
  const int lane = tid & 31;
  const int wave = tid >> 5;

  for (int idx = tid; idx < NHEAD * DIM; idx += 256) sW[idx] = W2e[idx];
  for (int idx = tid; idx < DIM; idx += 256) { sg[idx] = lng[idx]; sb[idx] = lnb[idx]; }
  if (i >= 1)
    for (int idx = tid; idx < DIM; idx += 256)
      sa[idx] = Ab[((size_t)b * SS + (i - 1)) * DIM + idx];
  if (tid < NHEAD) ssc[tid][0] = -3.0e38f;   // column 0 always masked
  __syncthreads();

  const float abias = attn_b[0];

  // each wave handles 8 of the 64 columns j = 1..64
  for (int t = 0; t < 8; ++t) {
    const int j = 1 + wave + 8 * t;
    if (i >= 1 && j == i) {              // diagonal masked
      if (lane == 0)
        for (int h = 0; h < NHEAD; ++h) ssc[h][j] = -3.0e38f;
      continue;
    }
    const float* brow = Bb + ((size_t)b * SS + (j - 1)) * DIM;
    const float* arow = (i >= 1) ? (const float*)sa
                                 : (Ab + ((size_t)b * SS + (j - 1)) * DIM);
    float xv[DIM / 32];
    float s1 = 0.f, s2 = 0.f;
#pragma unroll 4
    for (int t2 = 0; t2 < DIM / 32; ++t2) {
      const int d = lane + 32 * t2;
      const float x = arow[d] + brow[d];
      xv[t2] = x; s1 += x; s2 += x * x;
    }
    s1 = wred_sum(s1);
    s2 = wred_sum(s2);
    const float mean = s1 * (1.0f / DIM);
    const float var  = s2 * (1.0f / DIM) - mean * mean;
    const float rstd = rsqrtf(var + 1e-5f);

    float dot[NHEAD];
#pragma unroll
    for (int h = 0; h < NHEAD; ++h) dot[h] = 0.f;
#pragma unroll 2
    for (int t2 = 0; t2 < DIM / 32; ++t2) {
      const int d = lane + 32 * t2;
      float y = (xv[t2] - mean) * rstd * sg[d] + sb[d];
      y = fmaxf(y, 0.f);
#pragma unroll
      for (int h = 0; h < NHEAD; ++h) dot[h] += y * sW[h * DIM + d];
    }
#pragma unroll
    for (int h = 0; h < NHEAD; ++h) {
      const float dh = wred_sum(dot[h]);
      if (lane == 0)
        ssc[h][j] = qs[((size_t)b * NHEAD + h) * NNS + i]
                  + ks[((size_t)b * NHEAD + h) * NNS + j]
                  + dh + bias2[h] + abias;
    }
  }
  __syncthreads();

  // softmax per head over j (masked entries underflow exactly to 0)
  for (int h = wave; h < NHEAD; h += 8) {
    float m = -3.0e38f;
    for (int j = lane; j < NNS; j += 32) m = fmaxf(m, ssc[h][j]);
    m = wred_max(m);
    float sum = 0.f;
    for (int j = lane; j < NNS; j += 32) sum += expf(ssc[h][j] - m);
    sum = wred_sum(sum);
    const float inv = 1.0f / sum;
    for (int j = lane; j < NNS; j += 32) {
      const float p = expf(ssc[h][j] - m) * inv;
      sp[h][j] = p;
      attn_out[(((size_t)b * NHEAD + h) * NNS + i) * NNS + j] = p;
    }
  }
  __syncthreads();

  // context[b,i,d] = sum_j p[h(d)][j] * V[b,j,d]
#pragma unroll
  for (int t2 = 0; t2 < 3; ++t2) {
    const int d = tid + 256 * t2;
    const int h = d >> 6;
    float acc = 0.f;
    for (int j = 0; j < NNS; ++j)
      acc += sp[h][j] * Vb[((size_t)b * NNS + j) * DIM + d];
    ctx[((size_t)b * NNS + i) * DIM + d] = acc;
  }
}

// ---------------------------------------------------------------------------
extern "C" void kernel_launch(void* const* d_in, const int* in_sizes, int n_in,
                              void* d_out, int out_size, void* d_ws, size_t ws_size,
                              hipStream_t stream)
{
  (void)in_sizes; (void)n_in; (void)out_size; (void)ws_size;
  const float* desc = (const float*)d_in[0];   // (32,64,768)
  const float* nve  = (const float*)d_in[1];   // (32,65,768)
  const float* w1   = (const float*)d_in[2];   // (768,1536)
  const float* b1   = (const float*)d_in[3];
  const float* lng  = (const float*)d_in[4];
  const float* lnb  = (const float*)d_in[5];
  const float* w2   = (const float*)d_in[6];   // (768,768)
  const float* b2   = (const float*)d_in[7];
  const float* qw   = (const float*)d_in[8];
  const float* qb   = (const float*)d_in[9];
  const float* kw   = (const float*)d_in[10];
  const float* kb   = (const float*)d_in[11];
  const float* vw   = (const float*)d_in[12];
  const float* vbi  = (const float*)d_in[13];
  const float* aw   = (const float*)d_in[14];  // (1,192)
  const float* ab   = (const float*)d_in[15];  // (1,)
  const float* ow   = (const float*)d_in[16];
  const float* ob   = (const float*)d_in[17];

  float* ws = (float*)d_ws;
  size_t o = 0;
  float* Ab    = ws + o; o += (size_t)BB * SS * DIM;    // W1a.de + b1
  float* Bb    = ws + o; o += (size_t)BB * SS * DIM;    // W1b.de
  float* Qb    = ws + o; o += (size_t)BB * NNS * DIM;
  float* Kb    = ws + o; o += (size_t)BB * NNS * DIM;
  float* Vb    = ws + o; o += (size_t)BB * NNS * DIM;
  float* Cx    = ws + o; o += (size_t)BB * NNS * DIM;
  float* W2e   = ws + o; o += (size_t)NHEAD * DIM;
  float* bias2 = ws + o; o += 16;
  float* qs    = ws + o; o += (size_t)BB * NHEAD * NNS;
  float* ks    = ws + o; o += (size_t)BB * NHEAD * NNS;
  // total ~9.6M floats (~38.4 MB) of workspace

  float* out_f    = (float*)d_out;                       // (32,65,768)
  float* attn_f   = out_f + (size_t)BB * NNS * DIM;      // (32,12,65,65)

  const dim3 blk(256);
  const dim3 gA(16, 12);   // M=2048 rows
  const dim3 gQ(17, 12);   // M=2080 rows

  // Per-node projections (all f32 WMMA GEMMs)
  gemm_wmma_f32<<<gA, blk, 0, stream>>>(desc, w1,        b1,      Ab, BB * SS, 2 * DIM);
  gemm_wmma_f32<<<gA, blk, 0, stream>>>(desc, w1 + DIM,  nullptr, Bb, BB * SS, 2 * DIM);
  gemm_wmma_f32<<<gQ, blk, 0, stream>>>(nve,  qw,        qb,      Qb, BB * NNS, DIM);
  gemm_wmma_f32<<<gQ, blk, 0, stream>>>(nve,  kw,        kb,      Kb, BB * NNS, DIM);
  gemm_wmma_f32<<<gQ, blk, 0, stream>>>(nve,  vw,        vbi,     Vb, BB * NNS, DIM);

  prep_w2e<<<dim3((NHEAD * DIM + 255) / 256), blk, 0, stream>>>(w2, b2, aw, W2e, bias2);
  qk_scores<<<dim3((BB * NHEAD * NNS + 255) / 256), blk, 0, stream>>>(Qb, Kb, aw, qs, ks);

  edge_attn_ctx<<<dim3(NNS, BB), blk, 0, stream>>>(Ab, Bb, Vb, lng, lnb, W2e, bias2,
                                                   qs, ks, ab, attn_f, Cx);

  // output = context @ out_w^T + out_b  -> first segment of d_out
  gemm_wmma_f32<<<gQ, blk, 0, stream>>>(Cx, ow, ob, out_f, BB * NNS, DIM);
}